// GridInterpolator_83202106458168
// MI455X (gfx1250) — compile-verified
//
#include <hip/hip_runtime.h>

// Trilinear grid interpolation, tuned for MI455X (gfx1250, wave32).
// Memory-bound gather: ~620 MB traffic, ~270 MFLOP -> optimize the VMEM path.
//
// Layout: 4 lanes cooperate on one point; lane r handles features [4r, 4r+4).
// Each corner fetch = 4 lanes x global_load_b128 at consecutive 16B offsets
// = one coalesced 64B request (the exact voxel record size).

#define NUM_GRIDS 8
#define GDIM      96
#define NFEAT     16
#define GRID_VOL  (GDIM * GDIM * GDIM)           // 884736 voxels per grid
#define SX        (GDIM * GDIM)                  // x stride in voxels
#define SY        (GDIM)                         // y stride in voxels

typedef __attribute__((ext_vector_type(4))) float v4f;

__global__ __launch_bounds__(256) void grid_interp_kernel(
    const float* __restrict__ vox,   // [8,96,96,96,16] fp32
    const int*   __restrict__ gidx,  // [N,1] int32
    const float* __restrict__ pts,   // [N,3] fp32 in [-1,1]
    float*       __restrict__ out,   // [N,16] fp32
    int n)
{
    const int tid   = blockIdx.x * blockDim.x + threadIdx.x;
    const int point = tid >> 2;   // 4 lanes per point (8 points per wave32)
    const int sub   = tid & 3;    // which float4 of the 16 features
    if (point >= n) return;

    // Streamed-once inputs: non-temporal so they don't evict voxel-table
    // lines from the 192MB L2 (table is ~42% L2-resident).
    const float px = __builtin_nontemporal_load(pts + (size_t)point * 3 + 0);
    const float py = __builtin_nontemporal_load(pts + (size_t)point * 3 + 1);
    const float pz = __builtin_nontemporal_load(pts + (size_t)point * 3 + 2);
    const int   g  = __builtin_nontemporal_load(gidx + point);

    // Remap [-1,1] -> [0,1]
    const float ux = (px + 1.0f) * 0.5f;
    const float uy = (py + 1.0f) * 0.5f;
    const float uz = (pz + 1.0f) * 0.5f;
    const bool inside = (ux >= 0.0f) & (ux <= 1.0f) &
                        (uy >= 0.0f) & (uy <= 1.0f) &
                        (uz >= 0.0f) & (uz <= 1.0f);

    // Scale to voxel space; truncation matches astype(int32) in the reference.
    const float sx = ux * (float)(GDIM - 1);
    const float sy = uy * (float)(GDIM - 1);
    const float sz = uz * (float)(GDIM - 1);
    const int bx = (int)sx, by = (int)sy, bz = (int)sz;
    const float fx = sx - (float)bx;
    const float fy = sy - (float)by;
    const float fz = sz - (float)bz;

    // Corner coords, clipped to the grid (matches jnp.clip(base+off, 0, G-1)).
    const int x0 = min(max(bx,     0), GDIM - 1);
    const int x1 = min(max(bx + 1, 0), GDIM - 1);
    const int y0 = min(max(by,     0), GDIM - 1);
    const int y1 = min(max(by + 1, 0), GDIM - 1);
    const int z0 = min(max(bz,     0), GDIM - 1);
    const int z1 = min(max(bz + 1, 0), GDIM - 1);

    const float* gbase = vox + (size_t)g * ((size_t)GRID_VOL * NFEAT) + (size_t)(sub * 4);

    // 8 independent b128 gathers in flight, then weighted accumulate.
    v4f acc = (v4f)0.0f;
#pragma unroll
    for (int k = 0; k < 8; ++k) {
        const int cx = (k & 1) ? x1 : x0;   // bit i of corner k toggles dim i
        const int cy = (k & 2) ? y1 : y0;
        const int cz = (k & 4) ? z1 : z0;
        const float w = ((k & 1) ? fx : 1.0f - fx) *
                        ((k & 2) ? fy : 1.0f - fy) *
                        ((k & 4) ? fz : 1.0f - fz);
        const size_t off = ((size_t)cx * SX + (size_t)cy * SY + (size_t)cz) * NFEAT;
        const v4f v = *reinterpret_cast<const v4f*>(gbase + off);  // global_load_b128
        acc += v * w;
    }

    const float m = inside ? 1.0f : 0.0f;
    acc *= m;

    // Coalesced write-once output: non-temporal store (th:TH_NT) to keep L2
    // capacity for the voxel table.
    v4f* op = reinterpret_cast<v4f*>(out) + ((size_t)point * 4 + sub);
    __builtin_nontemporal_store(acc, op);
}

extern "C" void kernel_launch(void* const* d_in, const int* in_sizes, int n_in,
                              void* d_out, int out_size, void* d_ws, size_t ws_size,
                              hipStream_t stream) {
    const float* vox  = (const float*)d_in[0];  // voxel_embeddings [8,96,96,96,16]
    const int*   gidx = (const int*)d_in[1];    // grid_indexes [N,1]
    const float* pts  = (const float*)d_in[2];  // points [N,3]
    float*       out  = (float*)d_out;          // [N,16]

    const int n = in_sizes[1];                  // N (grid_indexes flat count)
    const long long threads = (long long)n * 4; // 4 lanes per point
    const int block = 256;                      // 8 waves (wave32)
    const int grid  = (int)((threads + block - 1) / block);

    grid_interp_kernel<<<grid, block, 0, stream>>>(vox, gidx, pts, out, n);
}